// FeatureVarianceLoss_17205638988057
// MI455X (gfx1250) — compile-verified
//
#include <hip/hip_runtime.h>
#include <cmath>

// FeatureVarianceLoss on MI455X (gfx1250, wave32).
//
// Reformulated around the per-sample 16x16 Gram matrix G = X X^T (X: 16x256 f32),
// computed with V_WMMA_F32_16X16X4_F32 accumulated over K=256 (64 steps).
// Everything else (norms, pair_sum, log-L1) falls out of G. Single pass over
// the 32 MB input -> bandwidth bound (~1.4us at 23.3 TB/s).

typedef __attribute__((ext_vector_type(2))) float v2f;
typedef __attribute__((ext_vector_type(8))) float v8f;

#define WAVES_PER_BLOCK 4
#define VV 16
#define CC 256

__global__ __launch_bounds__(128) void fvl_gram_kernel(
    const float* __restrict__ desc,    // [N,16,256]
    const float* __restrict__ vpred,   // [N,16]
    float* __restrict__ per_sample,    // [N] losses
    int N)
{
    __shared__ float gmat[WAVES_PER_BLOCK][16][17];  // padded: conflict-free column reads
    __shared__ float ginv[WAVES_PER_BLOCK][16];

    const int lane = threadIdx.x & 31;
    const int wave = threadIdx.x >> 5;
    const int n    = blockIdx.x * WAVES_PER_BLOCK + wave;
    const bool active = (n < N);
    const int nc = active ? n : 0;   // clamp so inactive waves stay in-bounds

    // ---- G = X X^T via v_wmma_f32_16x16x4_f32, K swept in steps of 4 ----
    // A-layout (32-bit 16x4): lanes 0-15 hold {K=k, K=k+1} for row M=lane,
    // lanes 16-31 hold {K=k+2, K=k+3} for row M=lane-16. For the symmetric
    // product the same registers serve as both A and B.
    v8f acc = {0.f, 0.f, 0.f, 0.f, 0.f, 0.f, 0.f, 0.f};

    const int row  = lane & 15;
    const int koff = (lane >> 4) << 1;  // 0 for lanes 0-15, 2 for lanes 16-31
    const float* base = desc + (size_t)nc * (VV * CC) + row * CC + koff;

    #pragma unroll 8
    for (int k = 0; k < CC; k += 4) {
        v2f ab = *(const v2f*)(base + k);   // 8B load, 8B-aligned
        acc = __builtin_amdgcn_wmma_f32_16x16x4_f32(
            /*neg_a=*/false, ab, /*neg_b=*/false, ab,
            /*c_mod=*/(short)0, acc, /*reuse_a=*/false, /*reuse_b=*/false);
    }

    // ---- spill G to LDS: C/D layout = VGPR r holds rows r (lanes 0-15) / r+8 ----
    const int rbase = (lane >> 4) << 3;  // 0 or 8
    #pragma unroll
    for (int r = 0; r < 8; ++r)
        gmat[wave][rbase + r][lane & 15] = acc[r];
    __syncthreads();

    // ---- inv_i = 1 / max(sqrt(G_ii), 1e-12)  (F.normalize eps) ----
    if (lane < 16) {
        float gii = gmat[wave][lane][lane];
        ginv[wave][lane] = 1.0f / fmaxf(sqrtf(gii), 1e-12f);
    }
    __syncthreads();

    // ---- sq = sum_i G_ii inv_i^2 ; S = sum_ij G_ij inv_i inv_j ; vp mean ----
    float s_part = 0.f, sq_part = 0.f, vp_part = 0.f;
    if (lane < 16) {
        const float invj = ginv[wave][lane];
        float cs = 0.f;
        #pragma unroll
        for (int i = 0; i < 16; ++i)
            cs += gmat[wave][i][lane] * ginv[wave][i];
        s_part  = cs * invj;
        sq_part = gmat[wave][lane][lane] * invj * invj;
        vp_part = vpred[nc * VV + lane];
    }
    #pragma unroll
    for (int m = 16; m >= 1; m >>= 1) {      // lanes 16-31 contribute zeros
        s_part  += __shfl_xor(s_part,  m, 32);
        sq_part += __shfl_xor(sq_part, m, 32);
        vp_part += __shfl_xor(vp_part, m, 32);
    }

    if (active && lane == 0) {
        float pair_sum = (float)VV * sq_part - s_part;     // V*sq - |s|^2
        float var_gt   = fmaxf(pair_sum * (1.0f / 120.0f), 0.05f);  // /C(16,2), clamp
        float vp       = vp_part * (1.0f / (float)VV) + 1e-6f;
        per_sample[n]  = fabsf(logf(vp) - logf(var_gt));
    }
}

// Deterministic single-block mean reduction (no float atomics).
__global__ __launch_bounds__(256) void fvl_reduce_kernel(
    const float* __restrict__ per_sample, float* __restrict__ out, int N)
{
    __shared__ float sdata[256];
    float s = 0.f;
    for (int i = threadIdx.x; i < N; i += 256)
        s += per_sample[i];
    sdata[threadIdx.x] = s;
    __syncthreads();
    #pragma unroll
    for (int stride = 128; stride > 0; stride >>= 1) {
        if (threadIdx.x < (unsigned)stride)
            sdata[threadIdx.x] += sdata[threadIdx.x + stride];
        __syncthreads();
    }
    if (threadIdx.x == 0)
        out[0] = sdata[0] / (float)N;
}

extern "C" void kernel_launch(void* const* d_in, const int* in_sizes, int n_in,
                              void* d_out, int out_size, void* d_ws, size_t ws_size,
                              hipStream_t stream)
{
    const float* desc  = (const float*)d_in[0];   // [N,16,256] f32
    const float* vpred = (const float*)d_in[1];   // [N,16,1]   f32
    float* out = (float*)d_out;
    float* per_sample = (float*)d_ws;             // N floats of scratch

    const int N = in_sizes[0] / (VV * CC);        // 8192
    const int blocks = (N + WAVES_PER_BLOCK - 1) / WAVES_PER_BLOCK;

    fvl_gram_kernel<<<blocks, 128, 0, stream>>>(desc, vpred, per_sample, N);
    fvl_reduce_kernel<<<1, 256, 0, stream>>>(per_sample, out, N);
}